// Loss_76905684402669
// MI455X (gfx1250) — compile-verified
//
#include <hip/hip_runtime.h>

// ---------------- problem constants (fixed by reference) ----------------
#define UNUM    128
#define NITEMS  512
#define NF      80
#define FEAT    64      // FIRST_USER_FEATURE
#define TOPK    10
#define NTHREADS 512
#define NWAVES  16
#define INV_TAU 10.0f   // TAU_NDCG == TAU_DIV == 0.1

typedef __attribute__((ext_vector_type(2))) float v2f;
typedef __attribute__((ext_vector_type(8))) float v8f;

__device__ __forceinline__ float waveSum(float v) {
#pragma unroll
  for (int o = 16; o > 0; o >>= 1) v += __shfl_xor(v, o, 32);
  return v;
}

__device__ __forceinline__ void waveArgMax(float& v, int& idx) {
#pragma unroll
  for (int o = 16; o > 0; o >>= 1) {
    float ov = __shfl_xor(v, o, 32);
    int   oi = __shfl_xor(idx, o, 32);
    if (ov > v || (ov == v && oi < idx)) { v = ov; idx = oi; }
  }
}

// block-wide sum over 512 threads (16 waves); scr needs >= 16 floats
__device__ __forceinline__ float blockSum(float v, float* scr, int tid) {
  v = waveSum(v);
  if ((tid & 31) == 0) scr[tid >> 5] = v;
  __syncthreads();
  float r = (tid < NWAVES) ? scr[tid] : 0.0f;
  r = waveSum(r);
  if (tid == 0) scr[0] = r;
  __syncthreads();
  float res = scr[0];
  __syncthreads();
  return res;
}

__global__ __launch_bounds__(NTHREADS)
void ranker_fused_kernel(const float* __restrict__ batch,
                         const float* __restrict__ score_div,
                         const float* __restrict__ score_ndcg,
                         const float* __restrict__ rel,
                         float* __restrict__ out) {
  const int u    = blockIdx.x;
  const int br   = blockIdx.y;          // 0: ndcg branch, 1: diversity branch
  const int tid  = threadIdx.x;
  const int lane = tid & 31;
  const int wid  = tid >> 5;
  const float NEG_INF = -__builtin_inff();

  // LDS state: P[u] = diag(exp(lr)) * exp(v_i*q_j + bt_j) * diag(exp(lc))
  __shared__ float q [NITEMS];   // s_j / tau
  __shared__ float bt[NITEMS];   // -a_j / tau
  __shared__ float lr[NITEMS];   // log row scale
  __shared__ float lc[NITEMS];   // log col scale
  __shared__ float rl[NITEMS];   // relevance
  __shared__ float sc[NITEMS];   // selection scratch
  // W-matrix for WMMA A fragments: [0,512)=ar, [512,1024)=tf, [1024,1538)=0
  __shared__ float wz[3 * NITEMS + 2];
  __shared__ float invn[NITEMS]; // 1/max(||feat||,1e-8)
  __shared__ float nn [NITEMS];  // ||n_j||^2 (exact, handles clip)
  __shared__ float arel[TOPK];
  __shared__ float red[64];

  const float* s = (br == 0 ? score_ndcg : score_div) + (size_t)u * NITEMS;
  q[tid]  = s[tid] * INV_TAU;
  rl[tid] = rel[(size_t)u * NITEMS + tid];
  __syncthreads();

  // ---- a_j/tau = sum_k |q_j - q_k|  (|s_j-s_k|/tau) ----
  {
    float qj = q[tid], acc = 0.0f;
    for (int k = 0; k < NITEMS; ++k) acc += fabsf(qj - q[k]);
    bt[tid] = -acc;
  }
  __syncthreads();

  // ---- init lr_i = -LSE_j(v_i*q_j + bt_j)  (== row softmax, exact) ----
  {
    float vi = (float)(NITEMS - 1 - 2 * tid);
    float m = NEG_INF, sum = 0.0f;
    for (int j = 0; j < NITEMS; ++j) {
      float x = fmaf(vi, q[j], bt[j]);
      if (x > m) { sum = fmaf(sum, expf(m - x), 1.0f); m = x; }
      else sum += expf(x - m);
    }
    lr[tid] = -(m + logf(sum));
  }
  __syncthreads();

  // ---- Sinkhorn: scaling vectors only (matrix is rank-structured) ----
  for (int it = 0; it < 5; ++it) {
    { // column normalize: lc_j = -(bt_j + LSE_i(lr_i + v_i*q_j))
      float qj = q[tid];
      float m = NEG_INF, sum = 0.0f;
      for (int i = 0; i < NITEMS; ++i) {
        float x = fmaf((float)(NITEMS - 1 - 2 * i), qj, lr[i]);
        if (x > m) { sum = fmaf(sum, expf(m - x), 1.0f); m = x; }
        else sum += expf(x - m);
      }
      lc[tid] = -(bt[tid] + m + logf(sum));
    }
    __syncthreads();
    { // row normalize: lr_i = -LSE_j(v_i*q_j + bt_j + lc_j)
      float vi = (float)(NITEMS - 1 - 2 * tid);
      float m = NEG_INF, sum = 0.0f;
      for (int j = 0; j < NITEMS; ++j) {
        float x = fmaf(vi, q[j], bt[j] + lc[j]);
        if (x > m) { sum = fmaf(sum, expf(m - x), 1.0f); m = x; }
        else sum += expf(x - m);
      }
      lr[tid] = -(m + logf(sum));
    }
    __syncthreads();
  }

  if (br == 0) {
    // ================= NDCG branch =================
    if (wid < TOPK) {             // approx_rel rows 0..9 only (TOP_K)
      int i = wid;
      float vi = (float)(NITEMS - 1 - 2 * i), lri = lr[i], acc = 0.0f;
      for (int j = lane; j < NITEMS; j += 32)
        acc += rl[j] * expf(lri + fmaf(vi, q[j], bt[j] + lc[j]));
      acc = waveSum(acc);
      if (lane == 0) arel[i] = acc;
    }
    sc[tid] = q[tid];             // selection copy of score (monotone in s)
    __syncthreads();

    if (wid == 0) {               // wave-synchronous top-K selections
      volatile float* vsc = sc;
      float dcg = 0.0f;
      for (int r = 0; r < TOPK; ++r) {
        float best = NEG_INF; int bi = NITEMS;
        for (int j = lane; j < NITEMS; j += 32) {
          float x = vsc[j];
          if (x > best || (x == best && j < bi)) { best = x; bi = j; }
        }
        waveArgMax(best, bi);
        if (lane == 0) vsc[bi] = NEG_INF;
        dcg += (exp2f(rl[bi]) - 1.0f) / log2f((float)(r + 2));
      }
      for (int j = lane; j < NITEMS; j += 32) vsc[j] = rl[j];
      float bdcg = 0.0f;
      for (int r = 0; r < TOPK; ++r) {
        float best = NEG_INF; int bi = NITEMS;
        for (int j = lane; j < NITEMS; j += 32) {
          float x = vsc[j];
          if (x > best || (x == best && j < bi)) { best = x; bi = j; }
        }
        waveArgMax(best, bi);
        if (lane == 0) vsc[bi] = NEG_INF;
        bdcg += (exp2f(best) - 1.0f) / log2f((float)(r + 2));
      }
      if (lane == 0) {
        float adcg = 0.0f;
        for (int i = 0; i < TOPK; ++i)
          adcg += (exp2f(arel[i]) - 1.0f) / log2f((float)(i + 2));
        out[u]        = adcg / bdcg;   // approx_ndcg
        out[UNUM + u] = dcg  / bdcg;   // ndcg_score
      }
    }
  } else {
    // ================= Diversity branch =================
    { // approx_rank_j = sum_i (N-i) * P_ij ; then sigmoid shift
      float qj = q[tid], cj = bt[tid] + lc[tid], acc = 0.0f;
      for (int i = 0; i < NITEMS; ++i) {
        float x = lr[i] + fmaf((float)(NITEMS - 1 - 2 * i), qj, cj);
        acc += (float)(NITEMS - i) * expf(x);
      }
      wz[tid]            = 1.0f / (1.0f + expf(-(acc - (float)(NITEMS - TOPK))));
      wz[NITEMS + tid]   = 0.0f;       // real top-K indicator, filled below
      wz[2 * NITEMS + tid] = 0.0f;     // zero region for A rows >= 2
      if (tid < 2) wz[3 * NITEMS + tid] = 0.0f;
      sc[tid] = q[tid];
    }
    { // feature inverse norms (exact ||n||^2 to honor the 1e-8 clip)
      const float* fb = batch + ((size_t)u * NITEMS + tid) * NF;
      __builtin_prefetch(fb, 0, 1);
      float ss = 0.0f;
      for (int f = 0; f < FEAT; ++f) { float x = fb[f]; ss = fmaf(x, x, ss); }
      float iv = 1.0f / fmaxf(sqrtf(ss), 1e-8f);
      invn[tid] = iv;
      nn[tid]   = ss * iv * iv;
    }
    __syncthreads();

    if (wid == 0) {               // real top-10 indicator via selection
      volatile float* vsc = sc;
      volatile float* vtf = wz + NITEMS;
      for (int r = 0; r < TOPK; ++r) {
        float best = NEG_INF; int bi = NITEMS;
        for (int j = lane; j < NITEMS; j += 32) {
          float x = vsc[j];
          if (x > best || (x == best && j < bi)) { best = x; bi = j; }
        }
        waveArgMax(best, bi);
        if (lane == 0) { vsc[bi] = NEG_INF; vtf[bi] = 1.0f; }
      }
    }
    __syncthreads();

    // WMMA GEMM: C(16x64) = W(16x512) * Nfeat(512x64), f32 16x16x4.
    // W row0 = ar, row1 = tf, rows 2..15 read the zero region of wz ->
    // every lane does one unconditional aligned ds_load_b64 per K-step
    // (no predicated loads / EXEC save-restore in the hot loop).
    float va2 = 0.0f, vr2 = 0.0f;
    if (wid < 4) {                          // 4 waves, one 16-col tile each
      v8f c = {0.f, 0.f, 0.f, 0.f, 0.f, 0.f, 0.f, 0.f};
      const int mrow  = lane & 15;          // A: M index (ISA 16x4 layout)
      const int khalf = (lane >> 4) << 1;   // K pair: {0,1} or {2,3}
      const int col   = (wid << 4) + (lane & 15);
      const int abase = (mrow <= 1) ? mrow * NITEMS : 2 * NITEMS;
      const float* fb = batch + (size_t)u * NITEMS * NF;
      for (int k0 = 0; k0 < NITEMS; k0 += 4) {
        int ka = k0 + khalf;
        v2f a, b;
        a.x = wz[abase + ka];
        a.y = wz[abase + ka + 1];
        b.x = fb[(size_t)ka * NF + col]       * invn[ka];
        b.y = fb[(size_t)(ka + 1) * NF + col] * invn[ka + 1];
        c = __builtin_amdgcn_wmma_f32_16x16x4_f32(false, a, false, b,
                                                  (short)0, c, false, false);
      }
      // row0 -> c[0] (lanes 0-15), row1 -> c[1]; lanes 16-31 hold zero rows 8/9
      va2 = waveSum(c[0] * c[0]);
      vr2 = waveSum(c[1] * c[1]);
    }
    if (wid < 4 && lane == 0) { red[wid] = va2; red[8 + wid] = vr2; }
    __syncthreads();

    float arj = wz[tid], tj = wz[NITEMS + tid], nnj = nn[tid];
    float S1  = blockSum(arj,             red + 16, tid);
    float S2  = blockSum(arj * arj,       red + 16, tid);
    float S2n = blockSum(arj * arj * nnj, red + 16, tid);
    float T1  = blockSum(tj,              red + 16, tid);
    float T2n = blockSum(tj * nnj,        red + 16, tid);

    if (tid == 0) {
      float Va2 = red[0] + red[1] + red[2] + red[3];
      float Vr2 = red[8] + red[9] + red[10] + red[11];
      float dena = 0.5f * (S1 * S1 - S2);
      float numa = dena - 0.5f * (Va2 - S2n);
      float denr = 0.5f * (T1 * T1 - T1);      // t^2 == t
      float numr = denr - 0.5f * (Vr2 - T2n);
      out[2 * UNUM + u] = numa / dena;          // approx_div
      out[3 * UNUM + u] = numr / denr;          // div_score
    }
  }
}

extern "C" void kernel_launch(void* const* d_in, const int* in_sizes, int n_in,
                              void* d_out, int out_size, void* d_ws, size_t ws_size,
                              hipStream_t stream) {
  (void)in_sizes; (void)n_in; (void)out_size; (void)d_ws; (void)ws_size;
  const float* batch = (const float*)d_in[0];
  const float* sdiv  = (const float*)d_in[1];
  const float* sndcg = (const float*)d_in[2];
  const float* rel   = (const float*)d_in[3];
  float* out = (float*)d_out;
  dim3 grid(UNUM, 2);
  ranker_fused_kernel<<<grid, NTHREADS, 0, stream>>>(batch, sdiv, sndcg, rel, out);
}